// NMS3dAndComposeA_73400991088752
// MI455X (gfx1250) — compile-verified
//
#include <hip/hip_runtime.h>
#include <hip/hip_bf16.h>
#include <stdint.h>

#define IMG_H 2048
#define IMG_W 2048
#define N_PIX (IMG_H * IMG_W)

constexpr int TX     = 64;        // output tile width  (multiple of 64 -> aligned chunks)
constexpr int TY     = 16;        // output tile height
constexpr int CHUNKS = 18;        // 18 x 4-float chunks = 72 staged columns
constexpr int LW     = CHUNKS*4;  // 72 staged cols; compute reads local cols 3..68
constexpr int LH     = TY + 2;    // 18 halo rows
constexpr int NTHR   = 256;       // 8 wave32s

__global__ __launch_bounds__(NTHR)
void nms3d_compose_kernel(const float* __restrict__ plow,
                          const float* __restrict__ pcur,
                          const float* __restrict__ phigh,
                          float* __restrict__ out)
{
    __shared__ __align__(16) float sm[3][LH][LW];
    const int tid = threadIdx.x;
    const int x0  = blockIdx.x * TX;
    const int y0  = blockIdx.y * TY;

    // ---- Stage 3-channel halo tile into LDS with CDNA5 async global->LDS b128 copies.
    // Staged window: rows [y0-1, y0+TY+1), cols [x0-4, x0+68) in 16B-aligned chunks.
    #pragma unroll
    for (int ch = 0; ch < 3; ++ch) {
        const float* base = (ch == 0) ? plow : (ch == 1) ? pcur : phigh;
        for (int i = tid; i < LH * CHUNKS; i += NTHR) {
            const int row = i / CHUNKS;
            const int k   = i - row * CHUNKS;
            int gy  = y0 - 1 + row;
            int cxs = x0 - 4 + 4 * k;
            gy  = gy  < 0 ? 0 : (gy  > IMG_H - 1 ? IMG_H - 1 : gy);   // clamp; OOB cells
            cxs = cxs < 0 ? 0 : (cxs > IMG_W - 4 ? IMG_W - 4 : cxs);  // re-zeroed below
            const int      goff = (gy * IMG_W + cxs) * 4;             // 16B-aligned byte off
            const uint32_t doff = (uint32_t)(uintptr_t)(&sm[ch][row][4 * k]);
            asm volatile("global_load_async_to_lds_b128 %0, %1, %2"
                         :: "v"(doff), "v"(goff), "s"(base)
                         : "memory");
        }
    }
    asm volatile("s_wait_asynccnt 0x0" ::: "memory");

    // Zero-fill cells outside the image (conv zero padding); only edge tiles hit this.
    if (x0 == 0 || y0 == 0 || x0 + TX == IMG_W || y0 + TY == IMG_H) {
        for (int i = tid; i < LH * LW; i += NTHR) {
            const int row = i / LW;
            const int col = i - row * LW;
            const int gy = y0 - 1 + row;
            const int gx = x0 - 4 + col;
            if (gy < 0 || gy >= IMG_H || gx < 0 || gx >= IMG_W) {
                sm[0][row][col] = 0.0f;
                sm[1][row][col] = 0.0f;
                sm[2][row][col] = 0.0f;
            }
        }
    }
    __syncthreads();

    // ---- 27-point stencil out of LDS. lanes -> consecutive columns (bank-conflict free).
    const int   tx   = tid & (TX - 1);   // 0..63
    const int   tyb  = tid >> 6;         // 0..3
    const int   gx   = x0 + tx;
    const float SCL  = 1.0f / 2048.0f;   // 1/min(H,W) == 1/H == 1/W here

    #pragma unroll
    for (int r = 0; r < TY; r += NTHR / TX) {
        const int ly = tyb + r;          // sm row of (gy-1)
        const int gy = y0 + ly;

        float v[3][3][3];
        #pragma unroll
        for (int ch = 0; ch < 3; ++ch)
            #pragma unroll
            for (int j = 0; j < 3; ++j)
                #pragma unroll
                for (int k = 0; k < 3; ++k)
                    v[ch][j][k] = sm[ch][ly + j][tx + 3 + k];   // local col of gx-1 is tx+3

        // row sums per channel + 3x3x3 max
        float rs[3][3];
        float m = v[0][0][0];
        #pragma unroll
        for (int ch = 0; ch < 3; ++ch)
            #pragma unroll
            for (int j = 0; j < 3; ++j) {
                rs[ch][j] = v[ch][j][0] + v[ch][j][1] + v[ch][j][2];
                m = fmaxf(m, fmaxf(v[ch][j][0], fmaxf(v[ch][j][1], v[ch][j][2])));
            }

        const float Bl  = rs[0][0] + rs[0][1] + rs[0][2];   // box sums per channel
        const float Bc  = rs[1][0] + rs[1][1] + rs[1][2];
        const float Bh  = rs[2][0] + rs[2][1] + rs[2][2];
        const float den = Bl + Bc + Bh;
        const float T0  = rs[0][0] + rs[1][0] + rs[2][0];   // cross-channel row sums
        const float T2  = rs[0][2] + rs[1][2] + rs[2][2];
        float Cc0 = 0.0f, Cc2 = 0.0f;                       // cross-channel col sums
        #pragma unroll
        for (int ch = 0; ch < 3; ++ch)
            #pragma unroll
            for (int j = 0; j < 3; ++j) {
                Cc0 += v[ch][j][0];
                Cc2 += v[ch][j][2];
            }

        // conv weights {-0.5, 0.5, 1.5} == 0.5 + {-1, 0, 1}
        const float hden = 0.5f * den;
        const float num0 = hden + (Bh  - Bl);    // z (scale) coordinate
        const float num1 = hden + (T2  - T0);    // y coordinate
        const float num2 = hden + (Cc2 - Cc0);   // x coordinate

        const float cc       = v[1][1][1];
        const bool  interior = (gx > 0) && (gx < IMG_W - 1) && (gy > 0) && (gy < IMG_H - 1);
        const float resp     = (interior && (cc - m + 1e-5f > 0.0f)) ? cc : 0.0f;

        const float inv = 1.0f / (den + 1e-8f);
        const float s0  = num0 * inv * SCL;
        const float s1  = (num1 * inv + (float)gy) * SCL;
        const float s2  = (num2 * inv + (float)gx) * SCL;
        const float msk = resp > 0.0f ? 1.0f : 0.0f;

        const int idx = gy * IMG_W + gx;
        out[idx] = resp;                                   // resp_flat

        // full_A[idx] = msk * [[s0, 0, s2], [0, s0, s1]]  (row-major, 8B aligned)
        float2* Ap = (float2*)(out + (size_t)N_PIX + 6 * (size_t)idx);
        Ap[0] = make_float2(s0 * msk, 0.0f);
        Ap[1] = make_float2(s2 * msk, 0.0f);
        Ap[2] = make_float2(s0 * msk, s1 * msk);
    }
}

extern "C" void kernel_launch(void* const* d_in, const int* in_sizes, int n_in,
                              void* d_out, int out_size, void* d_ws, size_t ws_size,
                              hipStream_t stream) {
    const float* plow  = (const float*)d_in[0];
    const float* pcur  = (const float*)d_in[1];
    const float* phigh = (const float*)d_in[2];
    float*       out   = (float*)d_out;
    dim3 grid(IMG_W / TX, IMG_H / TY);
    nms3d_compose_kernel<<<grid, NTHR, 0, stream>>>(plow, pcur, phigh, out);
}